// GAT_13056700580098
// MI455X (gfx1250) — compile-verified
//
#include <hip/hip_runtime.h>
#include <hip/hip_bf16.h>

typedef float v2f __attribute__((ext_vector_type(2)));
typedef float v8f __attribute__((ext_vector_type(8)));

#define N_NODES  50000
#define N_EDGES  1600000
#define IN_DIM   256
#define HID      128
#define N_CLS    40
#define N_CLS_P  48          // N_CLS padded to a multiple of 16
#define NEG_SLOPE 0.2f

// ---------------------------------------------------------------------------
// GEMM via V_WMMA_F32_16X16X4_F32:  D[M,N] = A[M,K] * W[Np,K]^T
// One wave computes NT adjacent 16x16 tiles (16 x NT*16 strip) so the A
// fragment is loaded once per k-step and reused by NT back-to-back wmmas.
// W must have >= gridDim.x*NT*16 valid rows (zero-padded if needed); the
// store is guarded by col < N. No divergence anywhere in the K-loop.
// A 16x4 f32 layout: lane L (m = L&15, half = L>>4) holds K = 2*half + {0,1}.
// B 4x16 f32 layout: lane L (n = L&15, half = L>>4) holds K = 2*half + {0,1}.
// C/D 16x16 f32: VGPR r: lanes0-15 -> (M=r, N=lane); lanes16-31 -> (M=r+8).
// ---------------------------------------------------------------------------
template <int NT>
__global__ __launch_bounds__(32)
void gat_gemm_wmma_f32(const float* __restrict__ A,  // [M,K] row-major
                       const float* __restrict__ W,  // [>=Np,K] row-major
                       float* __restrict__ D,        // [M,N] row-major
                       int M, int K, int N) {
    const int lane = threadIdx.x & 31;
    const int r15  = lane & 15;
    const int half = lane >> 4;
    const int m0   = blockIdx.y * 16;
    const int n0   = blockIdx.x * (NT * 16);

    const float* aptr = A + (size_t)(m0 + r15) * K + 2 * half;
    const float* wp[NT];
#pragma unroll
    for (int t = 0; t < NT; ++t)
        wp[t] = W + (size_t)(n0 + 16 * t + r15) * K + 2 * half;

    v8f acc[NT];
#pragma unroll
    for (int t = 0; t < NT; ++t) acc[t] = (v8f){};

    for (int k0 = 0; k0 < K; k0 += 4) {
        float2 av = *(const float2*)(aptr + k0);
        v2f a; a.x = av.x; a.y = av.y;
#pragma unroll
        for (int t = 0; t < NT; ++t) {
            float2 bv = *(const float2*)(wp[t] + k0);
            v2f b; b.x = bv.x; b.y = bv.y;
            acc[t] = __builtin_amdgcn_wmma_f32_16x16x4_f32(
                false, a, false, b, (short)0, acc[t], false, false);
        }
    }

#pragma unroll
    for (int t = 0; t < NT; ++t) {
        int col = n0 + 16 * t + r15;
        if (col < N) {
#pragma unroll
            for (int r = 0; r < 8; ++r) {
                int row = m0 + r + 8 * half;
                D[(size_t)row * N + col] = acc[t][r];
            }
        }
    }
}

// Zero-padded copy: Wp[r,c] = (r < Nrows) ? W[r,c] : 0, for r < NrowsPad
__global__ void gat_pad_w_kernel(const float* __restrict__ W, float* __restrict__ Wp,
                                 int Nrows, int NrowsPad, int K) {
    int i = blockIdx.x * blockDim.x + threadIdx.x;
    if (i >= NrowsPad * K) return;
    int r = i / K;
    Wp[i] = (r < Nrows) ? W[i] : 0.f;
}

// ---------------------------------------------------------------------------
// Per-node attention logits: as[n] = dot(h[n,:], a_src), ad[n] = dot(h[n,:], a_dst)
// One wave32 per node, shuffle reduction.
// ---------------------------------------------------------------------------
__global__ void gat_alpha_kernel(const float* __restrict__ h,
                                 const float* __restrict__ a_src,
                                 const float* __restrict__ a_dst,
                                 float* __restrict__ as_, float* __restrict__ ad_,
                                 int Nn, int C) {
    int wave = (blockIdx.x * blockDim.x + threadIdx.x) >> 5;
    int lane = threadIdx.x & 31;
    if (wave >= Nn) return;
    const float* hp = h + (size_t)wave * C;
    float s = 0.f, d = 0.f;
    for (int f = lane; f < C; f += 32) {
        float v = hp[f];
        s += v * a_src[f];
        d += v * a_dst[f];
    }
#pragma unroll
    for (int off = 16; off > 0; off >>= 1) {
        s += __shfl_xor(s, off, 32);
        d += __shfl_xor(d, off, 32);
    }
    if (lane == 0) { as_[wave] = s; ad_[wave] = d; }
}

__device__ __forceinline__ void edge_sd(const int* __restrict__ ei, int e,
                                        int E, int& s, int& d) {
    if (e < E) { s = ei[e]; d = ei[E + e]; }
    else       { s = e - E; d = e - E; }          // self loops appended
}

// Monotone encode of float for unsigned atomicMax
__device__ __forceinline__ unsigned f32_enc(float v) {
    unsigned u = __float_as_uint(v);
    return (u & 0x80000000u) ? ~u : (u | 0x80000000u);
}
__device__ __forceinline__ float f32_dec(unsigned k) {
    unsigned u = (k & 0x80000000u) ? (k & 0x7FFFFFFFu) : ~k;
    return __uint_as_float(u);
}

// Pass 1: e = leaky_relu(as[src] + ad[dst]); store e; segment max over dst
__global__ void gat_edge_max_kernel(const int* __restrict__ ei,
                                    const float* __restrict__ as_,
                                    const float* __restrict__ ad_,
                                    unsigned* __restrict__ m_enc,
                                    float* __restrict__ ebuf,
                                    int E, int Nn) {
    int e = blockIdx.x * blockDim.x + threadIdx.x;
    if (e >= E + Nn) return;
    int s, d; edge_sd(ei, e, E, s, d);
    float v = as_[s] + ad_[d];
    v = (v > 0.f) ? v : NEG_SLOPE * v;
    ebuf[e] = v;
    atomicMax(&m_enc[d], f32_enc(v));
}

// Pass 2: ex = exp(e - m[dst]); store ex; segment sum of ex over dst
__global__ void gat_edge_expsum_kernel(const int* __restrict__ ei,
                                       const unsigned* __restrict__ m_enc,
                                       float* __restrict__ ebuf,
                                       float* __restrict__ denom,
                                       int E, int Nn) {
    int e = blockIdx.x * blockDim.x + threadIdx.x;
    if (e >= E + Nn) return;
    int s, d; edge_sd(ei, e, E, s, d);
    float m  = f32_dec(m_enc[d]);
    float ex = expf(ebuf[e] - m);
    ebuf[e] = ex;
    atomicAdd(&denom[d], ex);
}

// Pass 3: out[dst,:] += (ex/denom[dst]) * h[src,:]   — one wave per edge
__global__ void gat_edge_agg_kernel(const int* __restrict__ ei,
                                    const float* __restrict__ h,
                                    const float* __restrict__ ebuf,
                                    const float* __restrict__ denom,
                                    float* __restrict__ out,
                                    int E, int Nn, int C) {
    int gid  = blockIdx.x * blockDim.x + threadIdx.x;
    int e    = gid >> 5;
    int lane = gid & 31;
    if (e >= E + Nn) return;
    int s, d; edge_sd(ei, e, E, s, d);
    float w = ebuf[e] / fmaxf(denom[d], 1e-16f);
    const float* hp = h + (size_t)s * C;
    float* op = out + (size_t)d * C;
    for (int f = lane; f < C; f += 32)
        atomicAdd(&op[f], hp[f] * w);
}

__global__ void gat_bias_act_kernel(float* __restrict__ x, const float* __restrict__ b,
                                    int total, int C, int do_relu) {
    int i = blockIdx.x * blockDim.x + threadIdx.x;
    if (i >= total) return;
    float v = x[i] + b[i % C];
    x[i] = do_relu ? fmaxf(v, 0.f) : v;
}

__global__ void gat_fill_u32_kernel(unsigned* __restrict__ p, unsigned v, size_t n) {
    size_t i = (size_t)blockIdx.x * blockDim.x + threadIdx.x;
    if (i < n) p[i] = v;
}

extern "C" void kernel_launch(void* const* d_in, const int* in_sizes, int n_in,
                              void* d_out, int out_size, void* d_ws, size_t ws_size,
                              hipStream_t stream) {
    const float* x    = (const float*)d_in[0];
    const int*   ei   = (const int*)  d_in[1];   // int32 (jax default x64 off)
    const float* W0   = (const float*)d_in[2];
    const float* a_s0 = (const float*)d_in[3];
    const float* a_d0 = (const float*)d_in[4];
    const float* b0   = (const float*)d_in[5];
    const float* W1   = (const float*)d_in[6];
    const float* a_s1 = (const float*)d_in[7];
    const float* a_d1 = (const float*)d_in[8];
    const float* b1   = (const float*)d_in[9];
    float* out = (float*)d_out;

    const int Nn = in_sizes[0] / IN_DIM;     // 50000
    const int E  = in_sizes[1] / 2;          // 1600000
    const int ET = E + Nn;                   // with self loops

    // workspace partition (floats)
    float* ws = (float*)d_ws;
    size_t off = 0;
    float*    h0    = ws + off; off += (size_t)Nn * HID;
    float*    out0  = ws + off; off += (size_t)Nn * HID;
    float*    h1    = ws + off; off += (size_t)Nn * N_CLS;
    float*    W1p   = ws + off; off += (size_t)N_CLS_P * HID;   // zero-padded W1
    float*    as_   = ws + off; off += Nn;
    float*    ad_   = ws + off; off += Nn;
    unsigned* m_enc = (unsigned*)(ws + off); off += Nn;
    float*    denom = ws + off; off += Nn;
    float*    ebuf  = ws + off; off += (size_t)ET;
    if (ws_size < off * sizeof(float)) return;   // scratch too small: bail deterministically

    const int BLK = 256;
    dim3 eGrid((ET + BLK - 1) / BLK);
    dim3 aGrid((unsigned)(((size_t)ET * 32 + BLK - 1) / BLK));
    dim3 nWaveGrid((Nn * 32 + BLK - 1) / BLK);

    // ---------------- Layer 0 ----------------
    // GEMM0: 16x64 strip per wave (NT=4), N=128 -> grid.x = 2
    gat_gemm_wmma_f32<4><<<dim3(HID / 64, Nn / 16), 32, 0, stream>>>(
        x, W0, h0, Nn, IN_DIM, HID);
    gat_alpha_kernel<<<nWaveGrid, BLK, 0, stream>>>(h0, a_s0, a_d0, as_, ad_, Nn, HID);

    gat_fill_u32_kernel<<<(Nn + BLK - 1) / BLK, BLK, 0, stream>>>(m_enc, 0u, (size_t)Nn);
    gat_fill_u32_kernel<<<(Nn + BLK - 1) / BLK, BLK, 0, stream>>>((unsigned*)denom, 0u, (size_t)Nn);
    gat_fill_u32_kernel<<<(unsigned)(((size_t)Nn * HID + BLK - 1) / BLK), BLK, 0, stream>>>(
        (unsigned*)out0, 0u, (size_t)Nn * HID);

    gat_edge_max_kernel   <<<eGrid, BLK, 0, stream>>>(ei, as_, ad_, m_enc, ebuf, E, Nn);
    gat_edge_expsum_kernel<<<eGrid, BLK, 0, stream>>>(ei, m_enc, ebuf, denom, E, Nn);
    gat_edge_agg_kernel   <<<aGrid, BLK, 0, stream>>>(ei, h0, ebuf, denom, out0, E, Nn, HID);
    gat_bias_act_kernel<<<(unsigned)(((size_t)Nn * HID + BLK - 1) / BLK), BLK, 0, stream>>>(
        out0, b0, Nn * HID, HID, /*relu=*/1);

    // ---------------- Layer 1 ----------------
    // Pad W1 [40,128] -> W1p [48,128] so the GEMM K-loop is divergence-free.
    gat_pad_w_kernel<<<(N_CLS_P * HID + BLK - 1) / BLK, BLK, 0, stream>>>(
        W1, W1p, N_CLS, N_CLS_P, HID);
    // GEMM1: 16x48 strip per wave (NT=3), padded N=48 -> grid.x = 1
    gat_gemm_wmma_f32<3><<<dim3(1, Nn / 16), 32, 0, stream>>>(
        out0, W1p, h1, Nn, HID, N_CLS);
    gat_alpha_kernel<<<nWaveGrid, BLK, 0, stream>>>(h1, a_s1, a_d1, as_, ad_, Nn, N_CLS);

    gat_fill_u32_kernel<<<(Nn + BLK - 1) / BLK, BLK, 0, stream>>>(m_enc, 0u, (size_t)Nn);
    gat_fill_u32_kernel<<<(Nn + BLK - 1) / BLK, BLK, 0, stream>>>((unsigned*)denom, 0u, (size_t)Nn);
    gat_fill_u32_kernel<<<(unsigned)(((size_t)Nn * N_CLS + BLK - 1) / BLK), BLK, 0, stream>>>(
        (unsigned*)out, 0u, (size_t)Nn * N_CLS);

    gat_edge_max_kernel   <<<eGrid, BLK, 0, stream>>>(ei, as_, ad_, m_enc, ebuf, E, Nn);
    gat_edge_expsum_kernel<<<eGrid, BLK, 0, stream>>>(ei, m_enc, ebuf, denom, E, Nn);
    gat_edge_agg_kernel   <<<aGrid, BLK, 0, stream>>>(ei, h1, ebuf, denom, out, E, Nn, N_CLS);
    gat_bias_act_kernel<<<(unsigned)(((size_t)Nn * N_CLS + BLK - 1) / BLK), BLK, 0, stream>>>(
        out, b1, Nn * N_CLS, N_CLS, /*relu=*/0);
}